// Reduction_Layer_12876311954034
// MI455X (gfx1250) — compile-verified
//
#include <hip/hip_runtime.h>
#include <math.h>

// ---------------------------------------------------------------------------
// Tropical (max-times) matmul: out[i,k] = max_j x[i,j] * sigmoid(A[j,k])
//   x: [2048, 256] f32, A: [256, 512] f32, out: [2048, 512] f32
//
// CDNA5 strategy:
//  * V_WMMA_F32_16X16X4_F32 computes exact fp32 outer products x[m]*w[n]
//    (A filled with x in all K slots, B with one non-zero K row), then
//    v_max_num_f32 folds each 16x16 product tile into the accumulator.
//  * x tiles staged via GLOBAL_LOAD_ASYNC_TO_LDS_B128 (+ s_wait_asynccnt).
//  * W tiles staged with fused sigmoid (expf) into LDS.
// ---------------------------------------------------------------------------

typedef __attribute__((ext_vector_type(2))) float v2f;
typedef __attribute__((ext_vector_type(8))) float v8f;
typedef __attribute__((ext_vector_type(4))) int   v4i;

typedef __attribute__((address_space(1))) v4i* gptr_v4i;
typedef __attribute__((address_space(3))) v4i* lptr_v4i;

#ifndef __has_builtin
#define __has_builtin(x) 0
#endif

#if __has_builtin(__builtin_amdgcn_wmma_f32_16x16x4_f32)
#define USE_WMMA 1
#else
#define USE_WMMA 0
#endif

#if __has_builtin(__builtin_amdgcn_global_load_async_to_lds_b128)
#define USE_ASYNC 1
#else
#define USE_ASYNC 0
#endif

__device__ __forceinline__ void wait_async_zero() {
#if USE_ASYNC
#if __has_builtin(__builtin_amdgcn_s_wait_asynccnt)
  __builtin_amdgcn_s_wait_asynccnt(0);
#else
  asm volatile("s_wait_asynccnt 0x0" ::: "memory");
#endif
#endif
}

constexpr int BATCH = 2048;
constexpr int KDIM  = 256;   // reduction dim J
constexpr int NDIM  = 512;   // output cols
constexpr int BM = 32;       // rows per block   (2 wave-rows of 16)
constexpr int BN = 64;       // cols per block   (4 wave-cols of 16)
constexpr int BK = 32;       // J-chunk per stage
constexpr int XPAD = 36;     // 144B row stride: 16B aligned, bank-conflict-free

__global__ __launch_bounds__(256)
void tropical_sigmoid_maxmul_kernel(const float* __restrict__ x,
                                    const float* __restrict__ A,
                                    float* __restrict__ out) {
  __shared__ float xs[BM][XPAD];   // x tile   [m][k]
  __shared__ float ws[BK][BN];     // sigmoid(A) tile [k][n]

  const int tid  = threadIdx.x;
  const int lane = tid & 31;
  const int l15  = lane & 15;
  const int half = lane >> 4;      // 0: M rows 0-7, 1: M rows 8-15
  const int wv   = tid >> 5;       // wave 0..7
  const int wr   = wv >> 2;        // wave row 0..1
  const int wc   = wv & 3;         // wave col 0..3

  const int row0 = blockIdx.y * BM;
  const int col0 = blockIdx.x * BN;

  // staging decomposition (256 threads)
  const int xm   = tid >> 3;        // 0..31: x-tile row
  const int xseg = tid & 7;         // 16-byte segment within the 128B row
  const int wk   = tid >> 3;        // 0..31: W-tile row
  const int wcol = (tid & 7) * 8;   // 8-float chunk within 64-float row

  v8f acc;
#pragma unroll
  for (int i = 0; i < 8; ++i) acc[i] = -__builtin_inff();
  const v8f czero = {};

  for (int kb = 0; kb < KDIM / BK; ++kb) {
    const int k0 = kb * BK;

    // ---- stage x tile: 4KB, one b128 async-to-LDS per thread ----
    const float* gx = x + (size_t)(row0 + xm) * KDIM + (k0 + xseg * 4);
#if USE_ASYNC
    __builtin_amdgcn_global_load_async_to_lds_b128(
        (gptr_v4i)gx,
        (lptr_v4i)&xs[xm][xseg * 4],
        0, 0);
#else
    *(float4*)&xs[xm][xseg * 4] = *(const float4*)gx;
#endif

    // ---- stage W tile with fused sigmoid (8 elements / thread) ----
    const float* ga = A + (size_t)(k0 + wk) * NDIM + (col0 + wcol);
#pragma unroll
    for (int c = 0; c < 8; c += 4) {
      const float4 a4 = *(const float4*)(ga + c);
      float4 s4;
      s4.x = 1.0f / (1.0f + expf(-a4.x));
      s4.y = 1.0f / (1.0f + expf(-a4.y));
      s4.z = 1.0f / (1.0f + expf(-a4.z));
      s4.w = 1.0f / (1.0f + expf(-a4.w));
      *(float4*)&ws[wk][wcol + c] = s4;
    }

    wait_async_zero();
    __syncthreads();

    // ---- compute: per j, one fp32 WMMA outer product + 8 v_max ----
#pragma unroll
    for (int k = 0; k < BK; ++k) {
      const float xv    = xs[wr * 16 + l15][k];        // x[m = l15]
      const float wfull = ws[k][wc * 16 + l15];        // w[n = l15]
#if USE_WMMA
      // A: x[m] replicated in both VGPR slots of every lane -> A[m][*] = x[m]
      // B: single non-zero K row (VGPR0, lanes 0-15) -> D[m][n] = x[m]*w[n]
      v2f a; a[0] = xv;  a[1] = xv;
      v2f b; b[0] = half ? 0.0f : wfull;  b[1] = 0.0f;
      v8f d = __builtin_amdgcn_wmma_f32_16x16x4_f32(
          false, a, false, b, (short)0, czero, false, false);
#pragma unroll
      for (int i = 0; i < 8; ++i) acc[i] = fmaxf(acc[i], d[i]);
#else
      // scalar fallback with identical output mapping
#pragma unroll
      for (int i = 0; i < 8; ++i)
        acc[i] = fmaxf(acc[i], xs[wr * 16 + half * 8 + i][k] * wfull);
#endif
    }
    __syncthreads();
  }

  // C/D layout: VGPR i -> (M = i + 8*half, N = l15)
  const int orow = row0 + wr * 16 + half * 8;
  const int ocol = col0 + wc * 16 + l15;
#pragma unroll
  for (int i = 0; i < 8; ++i)
    out[(size_t)(orow + i) * NDIM + ocol] = acc[i];
}

extern "C" void kernel_launch(void* const* d_in, const int* in_sizes, int n_in,
                              void* d_out, int out_size, void* d_ws, size_t ws_size,
                              hipStream_t stream) {
  (void)in_sizes; (void)n_in; (void)out_size; (void)d_ws; (void)ws_size;
  const float* x = (const float*)d_in[0];   // [2048, 256]
  const float* A = (const float*)d_in[1];   // [256, 512]
  float* out = (float*)d_out;               // [2048, 512]

  dim3 grid(NDIM / BN, BATCH / BM);         // (8, 64)
  tropical_sigmoid_maxmul_kernel<<<grid, dim3(256), 0, stream>>>(x, A, out);
}